// OptSE3_85899345920019
// MI455X (gfx1250) — compile-verified
//
#include <hip/hip_runtime.h>
#include <math.h>

typedef __attribute__((ext_vector_type(2))) float v2f;
typedef __attribute__((ext_vector_type(8))) float v8f;

// out(4xN) = M(4x4) @ x(4xN), M = se3_exp(w, v, theta).
// Each wave processes 32 points per iteration via two V_WMMA_F32_16X16X4_F32:
//   A = M replicated over 16 rows, B = 16-point chunk (4x16, column = point).
__global__ __launch_bounds__(256) void OptSE3_wmma_kernel(
    const float* __restrict__ x, const float* __restrict__ w,
    const float* __restrict__ v, const float* __restrict__ theta,
    float* __restrict__ out, long n)
{
    __shared__ float Ms[16];

    if (threadIdx.x == 0) {
        float wx = w[0], wy = w[1], wz = w[2];
        float vx = v[0], vy = v[1], vz = v[2];
        float th = theta[0];
        float s = sinf(th), c = cosf(th);
        float a  = 1.0f - c;      // coeff of KK in R, of K in V
        float bb = th - s;        // coeff of KK in V
        // KK = K*K (symmetric)
        float kk00 = -(wy * wy + wz * wz);
        float kk11 = -(wx * wx + wz * wz);
        float kk22 = -(wx * wx + wy * wy);
        float kk01 = wx * wy, kk02 = wx * wz, kk12 = wy * wz;
        // R = I + s*K + (1-c)*KK
        float R00 = 1.0f + a * kk00;
        float R01 = -s * wz + a * kk01;
        float R02 =  s * wy + a * kk02;
        float R10 =  s * wz + a * kk01;
        float R11 = 1.0f + a * kk11;
        float R12 = -s * wx + a * kk12;
        float R20 = -s * wy + a * kk02;
        float R21 =  s * wx + a * kk12;
        float R22 = 1.0f + a * kk22;
        // V = theta*I + (1-c)*K + (theta-s)*KK ; t = V @ v
        float V00 = th + bb * kk00;
        float V01 = -a * wz + bb * kk01;
        float V02 =  a * wy + bb * kk02;
        float V10 =  a * wz + bb * kk01;
        float V11 = th + bb * kk11;
        float V12 = -a * wx + bb * kk12;
        float V20 = -a * wy + bb * kk02;
        float V21 =  a * wx + bb * kk12;
        float V22 = th + bb * kk22;
        float tx = V00 * vx + V01 * vy + V02 * vz;
        float ty = V10 * vx + V11 * vy + V12 * vz;
        float tz = V20 * vx + V21 * vy + V22 * vz;
        Ms[0] = R00; Ms[1] = R01; Ms[2]  = R02; Ms[3]  = tx;
        Ms[4] = R10; Ms[5] = R11; Ms[6]  = R12; Ms[7]  = ty;
        Ms[8] = R20; Ms[9] = R21; Ms[10] = R22; Ms[11] = tz;
        Ms[12] = 0.0f; Ms[13] = 0.0f; Ms[14] = 0.0f; Ms[15] = 1.0f;
    }
    __syncthreads();

    const unsigned lane = threadIdx.x & 31u;
    const unsigned l16  = lane & 15u;
    const unsigned mrow = lane & 3u;              // A row (mod 4) for this lane
    const unsigned k0   = (lane >> 4) * 2u;       // K index pair held by this half-wave

    // A operand: 16x4 f32 layout — lanes 0-15 hold K=0,1; lanes 16-31 hold K=2,3.
    v2f amat;
    amat.x = Ms[mrow * 4u + k0];
    amat.y = Ms[mrow * 4u + k0 + 1u];

    const long r0 = (long)k0;                     // component row pair read by this half-wave
    long tid      = (long)blockIdx.x * blockDim.x + threadIdx.x;
    long waveBase = tid - (long)lane;             // 32-point span start, wave-uniform
    long stride   = (long)gridDim.x * blockDim.x; // points per grid iteration
    long nFull    = n & ~31L;

    for (long base = waveBase; base + 32 <= n; base += stride) {
        long c0 = base + (long)l16;
        long c1 = base + 16 + (long)l16;
        // B operand: 4x16 f32 — lanes 0-15 hold K=0,1; lanes 16-31 hold K=2,3.
        v2f b0, b1;
        b0.x = __builtin_nontemporal_load(&x[r0 * n + c0]);
        b0.y = __builtin_nontemporal_load(&x[(r0 + 1) * n + c0]);
        b1.x = __builtin_nontemporal_load(&x[r0 * n + c1]);
        b1.y = __builtin_nontemporal_load(&x[(r0 + 1) * n + c1]);

        v8f acc0 = {0.0f, 0.0f, 0.0f, 0.0f, 0.0f, 0.0f, 0.0f, 0.0f};
        v8f acc1 = {0.0f, 0.0f, 0.0f, 0.0f, 0.0f, 0.0f, 0.0f, 0.0f};
        v8f d0 = __builtin_amdgcn_wmma_f32_16x16x4_f32(
            false, amat, false, b0, (short)0, acc0, false, false);
        v8f d1 = __builtin_amdgcn_wmma_f32_16x16x4_f32(
            false, amat, false, b1, (short)0, acc1, false, false);

        // D VGPR j holds component j of point (lane&15) in every lane (A rows
        // replicated mod 4). Lanes 0-15 take chunk0, lanes 16-31 take chunk1:
        // the stored value for lane L is component j of global point base+L.
        bool lo = (lane < 16u);
        float o0 = lo ? d0[0] : d1[0];
        float o1 = lo ? d0[1] : d1[1];
        float o2 = lo ? d0[2] : d1[2];
        float o3 = lo ? d0[3] : d1[3];
        long p = base + (long)lane;
        __builtin_nontemporal_store(o0, &out[0 * n + p]);
        __builtin_nontemporal_store(o1, &out[1 * n + p]);
        __builtin_nontemporal_store(o2, &out[2 * n + p]);
        __builtin_nontemporal_store(o3, &out[3 * n + p]);
    }

    // Tail (n % 32 points) — plain VALU path, block 0 only.
    long rem = n - nFull;
    if (rem > 0 && blockIdx.x == 0 && (long)threadIdx.x < rem) {
        long p = nFull + (long)threadIdx.x;
        float x0 = x[p], x1 = x[n + p], x2 = x[2 * n + p], x3 = x[3 * n + p];
#pragma unroll
        for (int j = 0; j < 4; ++j) {
            out[j * n + p] = Ms[j * 4 + 0] * x0 + Ms[j * 4 + 1] * x1 +
                             Ms[j * 4 + 2] * x2 + Ms[j * 4 + 3] * x3;
        }
    }
}

extern "C" void kernel_launch(void* const* d_in, const int* in_sizes, int n_in,
                              void* d_out, int out_size, void* d_ws, size_t ws_size,
                              hipStream_t stream) {
    const float* x     = (const float*)d_in[0];
    const float* w     = (const float*)d_in[1];
    const float* v     = (const float*)d_in[2];
    const float* theta = (const float*)d_in[3];
    float* out         = (float*)d_out;

    long n = (long)in_sizes[0] / 4;   // number of points (columns)

    const int threads = 256;
    long needed = (n + threads - 1) / threads;
    int blocks = (int)(needed < 8192 ? (needed > 0 ? needed : 1) : 8192);

    OptSE3_wmma_kernel<<<blocks, threads, 0, stream>>>(x, w, v, theta, out, n);
}